// MultiModalTransformer_66580583023034
// MI455X (gfx1250) — compile-verified
//
#include <hip/hip_runtime.h>
#include <hip/hip_bf16.h>

typedef __attribute__((ext_vector_type(2))) float v2f;
typedef __attribute__((ext_vector_type(8))) float v8f;

// Workspace layout (float offsets)
#define WS_M    0      // 64x32 collapsed linear map  feats->h
#define WS_C    2048   // 64    collapsed bias
#define WS_M2   2112   // 64x32 BN-folded map
#define WS_C2   4160   // 64    BN-folded bias
#define WS_SH   4224   // 64    sum(h)
#define WS_SH2  4288   // 64    sum(h^2)

// ---------------------------------------------------------------------------
// Kernel 1: collapse the whole affine chain (qkv/attn/ab_proj/f1) into
//           M (64x32), c (64).  Also zero the batch-stat accumulators.
// ---------------------------------------------------------------------------
__global__ __launch_bounds__(64) void prep_kernel(
    const float* __restrict__ ehr_qkv_w, const float* __restrict__ ehr_qkv_b,
    const float* __restrict__ bio_qkv_w, const float* __restrict__ bio_qkv_b,
    const float* __restrict__ attn_in_w, const float* __restrict__ attn_in_b,
    const float* __restrict__ attn_out_w, const float* __restrict__ attn_out_b,
    const float* __restrict__ ab_proj_w, const float* __restrict__ ab_proj_b,
    const float* __restrict__ f1_w, const float* __restrict__ f1_b,
    float* __restrict__ ws)
{
    __shared__ float P[16*16], pv[16], G1[16*16], g1[16], G2[16*16], g2[16];
    __shared__ float Q[16*32], qv[16], R[16*32], rv[16];
    const int j = threadIdx.x;
    ws[WS_SH  + j] = 0.f;
    ws[WS_SH2 + j] = 0.f;

    const float* Wv = attn_in_w + 32*16;   // attn_in_w[2E:3E]
    const float* bv = attn_in_b + 32;
    if (j < 16) {                          // P = O*Wv ; p = O*bv + ob
        for (int k = 0; k < 16; ++k) {
            float a = 0.f;
            for (int i = 0; i < 16; ++i) a += attn_out_w[j*16+i] * Wv[i*16+k];
            P[j*16+k] = a;
        }
        float b = attn_out_b[j];
        for (int i = 0; i < 16; ++i) b += attn_out_w[j*16+i] * bv[i];
        pv[j] = b;
    }
    __syncthreads();
    const float* Ae = ehr_qkv_w + 32*16; const float* ae = ehr_qkv_b + 32;
    const float* Ab = bio_qkv_w + 32*16; const float* ab = bio_qkv_b + 32;
    if (j < 16) {                          // G1 = P*Ae, G2 = P*Ab (+bias)
        for (int k = 0; k < 16; ++k) {
            float a1 = 0.f, a2 = 0.f;
            for (int i = 0; i < 16; ++i) {
                a1 += P[j*16+i]*Ae[i*16+k];
                a2 += P[j*16+i]*Ab[i*16+k];
            }
            G1[j*16+k] = a1; G2[j*16+k] = a2;
        }
        float b1 = pv[j], b2 = pv[j];
        for (int i = 0; i < 16; ++i) { b1 += P[j*16+i]*ae[i]; b2 += P[j*16+i]*ab[i]; }
        g1[j] = b1; g2[j] = b2;
    }
    __syncthreads();
    if (j < 16) {                          // Q = Wab*[G1 0; 0 G2], q
        for (int k = 0; k < 16; ++k) {
            float aq = 0.f, bq = 0.f;
            for (int i = 0; i < 16; ++i) {
                aq += ab_proj_w[j*32+i]    * G1[i*16+k];
                bq += ab_proj_w[j*32+16+i] * G2[i*16+k];
            }
            Q[j*32+k] = aq; Q[j*32+16+k] = bq;
        }
        float qq = ab_proj_b[j];
        for (int i = 0; i < 16; ++i)
            qq += ab_proj_w[j*32+i]*g1[i] + ab_proj_w[j*32+16+i]*g2[i];
        qv[j] = qq;
    }
    __syncthreads();
    if (j < 16) {                          // R = P*Q ; r = P*q + p
        for (int k = 0; k < 32; ++k) {
            float a = 0.f;
            for (int i = 0; i < 16; ++i) a += P[j*16+i]*Q[i*32+k];
            R[j*32+k] = a;
        }
        float rr = pv[j];
        for (int i = 0; i < 16; ++i) rr += P[j*16+i]*qv[i];
        rv[j] = rr;
    }
    __syncthreads();
    // M = f1_w[:, :16]*Q + f1_w[:, 16:]*R ; c likewise  (all 64 threads)
    for (int k = 0; k < 32; ++k) {
        float a = 0.f;
        for (int i = 0; i < 16; ++i)
            a += f1_w[j*32+i]*Q[i*32+k] + f1_w[j*32+16+i]*R[i*32+k];
        ws[WS_M + j*32 + k] = a;
    }
    float cc = f1_b[j];
    for (int i = 0; i < 16; ++i)
        cc += f1_w[j*32+i]*qv[i] + f1_w[j*32+16+i]*rv[i];
    ws[WS_C + j] = cc;
}

// ---------------------------------------------------------------------------
// Kernel 3: fold batch-norm into the linear map:  M2 = M*s, c2 = c*s + t
// ---------------------------------------------------------------------------
__global__ __launch_bounds__(64) void fold_kernel(
    const float* __restrict__ bn_g, const float* __restrict__ bn_b,
    float* __restrict__ ws, float invB)
{
    const int j = threadIdx.x;
    float mu  = ws[WS_SH  + j] * invB;
    float var = ws[WS_SH2 + j] * invB - mu*mu;
    float sc  = bn_g[j] * rsqrtf(var + 1e-5f);
    float sft = bn_b[j] - mu*sc;
    for (int k = 0; k < 32; ++k)
        ws[WS_M2 + j*32 + k] = ws[WS_M + j*32 + k] * sc;
    ws[WS_C2 + j] = ws[WS_C + j] * sc + sft;
}

// ---------------------------------------------------------------------------
// Pass 1: per 16-row tile compute h = feats(16x32) @ M^T (32x64) via
//         v_wmma_f32_16x16x4_f32, accumulate sum(h), sum(h^2).
//         A fragment built directly in ISA layout:
//           v0: lanes0-15 (M=lane, K=4s+0), lanes16-31 (M=lane-16, K=4s+2)
//           v1:                K=4s+1                       K=4s+3
//         B fragment assumed mirrored (N across lanes, K split low/high).
// ---------------------------------------------------------------------------
__global__ __launch_bounds__(256) void pass1_kernel(
    const float* __restrict__ x,
    const float* __restrict__ ehr_w, const float* __restrict__ ehr_b,
    const float* __restrict__ bio_w, const float* __restrict__ bio_b,
    float* __restrict__ ws, int ntiles)
{
    const int lane = threadIdx.x & 31;
    const int hl   = lane >> 4;          // half-wave
    const int ln   = lane & 15;          // row (A) / column (B,C)
    const int koff = hl * 2;
    const float* Mm = ws + WS_M;
    const float* cv = ws + WS_C;

    // per-lane feat weights: k = 4s+koff+t  (s<4 -> ehr branch, s>=4 -> bio)
    float wk[16], bk[16];
    #pragma unroll
    for (int s = 0; s < 8; ++s)
        #pragma unroll
        for (int t = 0; t < 2; ++t) {
            int k = 4*s + koff + t;
            wk[2*s+t] = (k < 16) ? ehr_w[k] : bio_w[k-16];
            bk[2*s+t] = (k < 16) ? ehr_b[k] : bio_b[k-16];
        }
    // B fragments for the 4 N-tiles x 8 K-steps
    v2f bf[4][8];
    #pragma unroll
    for (int nt = 0; nt < 4; ++nt)
        #pragma unroll
        for (int s = 0; s < 8; ++s) {
            int n = 16*nt + ln;
            v2f t = { Mm[n*32 + 4*s + koff], Mm[n*32 + 4*s + koff + 1] };
            bf[nt][s] = t;
        }
    float cb[4];
    #pragma unroll
    for (int nt = 0; nt < 4; ++nt) cb[nt] = cv[16*nt + ln];

    v8f sh[4] = {}, sh2[4] = {};

    const int wave = blockIdx.x * (blockDim.x >> 5) + (threadIdx.x >> 5);
    const int nw   = gridDim.x * (blockDim.x >> 5);
    for (int tile = wave; tile < ntiles; tile += nw) {
        const int row = tile*16 + ln;
        const float x0 = x[row*3 + 0];
        const float x2 = x[row*3 + 2];
        v2f af[8];
        #pragma unroll
        for (int s = 0; s < 8; ++s) {
            float xs = (s < 4) ? x0 : x2;
            float f0 = fmaxf(fmaf(xs, wk[2*s],   bk[2*s]),   0.f);
            float f1 = fmaxf(fmaf(xs, wk[2*s+1], bk[2*s+1]), 0.f);
            v2f t = { f0, f1 };
            af[s] = t;
        }
        #pragma unroll
        for (int nt = 0; nt < 4; ++nt) {
            v8f acc = {};
            #pragma unroll
            for (int s = 0; s < 8; ++s)
                acc = __builtin_amdgcn_wmma_f32_16x16x4_f32(
                    false, af[s], false, bf[nt][s], (short)0, acc, false, false);
            #pragma unroll
            for (int e = 0; e < 8; ++e) {
                float h = acc[e] + cb[nt];
                sh[nt][e]  += h;
                sh2[nt][e] += h*h;
            }
        }
    }
    // lanes l and l+16 hold the same feature column -> atomic combine
    #pragma unroll
    for (int nt = 0; nt < 4; ++nt) {
        float s1 = 0.f, s2 = 0.f;
        #pragma unroll
        for (int e = 0; e < 8; ++e) { s1 += sh[nt][e]; s2 += sh2[nt][e]; }
        atomicAdd(&ws[WS_SH  + 16*nt + ln], s1);
        atomicAdd(&ws[WS_SH2 + 16*nt + ln], s2);
    }
}

// ---------------------------------------------------------------------------
// Pass 2: same WMMA GEMM with BN-folded M2/c2, relu, then 64->3 projection
//         via wave-private LDS transpose of the h-tile.
// ---------------------------------------------------------------------------
__global__ __launch_bounds__(256) void pass2_kernel(
    const float* __restrict__ x,
    const float* __restrict__ ehr_w, const float* __restrict__ ehr_b,
    const float* __restrict__ bio_w, const float* __restrict__ bio_b,
    const float* __restrict__ f2_w, const float* __restrict__ f2_b,
    const float* __restrict__ ws, float* __restrict__ out, int ntiles)
{
    __shared__ float hbuf[8*16*64];   // 8 waves x (16 rows x 64 feats)
    __shared__ float f2w[3*64];
    __shared__ float f2b[3];
    for (int i = threadIdx.x; i < 192; i += blockDim.x) f2w[i] = f2_w[i];
    if (threadIdx.x < 3) f2b[threadIdx.x] = f2_b[threadIdx.x];
    __syncthreads();

    const int lane = threadIdx.x & 31;
    const int hl   = lane >> 4;
    const int ln   = lane & 15;
    const int koff = hl * 2;
    const int wid  = threadIdx.x >> 5;
    float* lw = hbuf + wid*16*64;
    const float* Mm = ws + WS_M2;
    const float* cv = ws + WS_C2;

    float wk[16], bk[16];
    #pragma unroll
    for (int s = 0; s < 8; ++s)
        #pragma unroll
        for (int t = 0; t < 2; ++t) {
            int k = 4*s + koff + t;
            wk[2*s+t] = (k < 16) ? ehr_w[k] : bio_w[k-16];
            bk[2*s+t] = (k < 16) ? ehr_b[k] : bio_b[k-16];
        }
    v2f bf[4][8];
    #pragma unroll
    for (int nt = 0; nt < 4; ++nt)
        #pragma unroll
        for (int s = 0; s < 8; ++s) {
            int n = 16*nt + ln;
            v2f t = { Mm[n*32 + 4*s + koff], Mm[n*32 + 4*s + koff + 1] };
            bf[nt][s] = t;
        }
    float cb[4];
    #pragma unroll
    for (int nt = 0; nt < 4; ++nt) cb[nt] = cv[16*nt + ln];

    const int wave = blockIdx.x * (blockDim.x >> 5) + (threadIdx.x >> 5);
    const int nw   = gridDim.x * (blockDim.x >> 5);
    for (int tile = wave; tile < ntiles; tile += nw) {
        const int row = tile*16 + ln;
        const float x0 = x[row*3 + 0];
        const float x2 = x[row*3 + 2];
        v2f af[8];
        #pragma unroll
        for (int s = 0; s < 8; ++s) {
            float xs = (s < 4) ? x0 : x2;
            float f0 = fmaxf(fmaf(xs, wk[2*s],   bk[2*s]),   0.f);
            float f1 = fmaxf(fmaf(xs, wk[2*s+1], bk[2*s+1]), 0.f);
            v2f t = { f0, f1 };
            af[s] = t;
        }
        #pragma unroll
        for (int nt = 0; nt < 4; ++nt) {
            v8f acc = {};
            #pragma unroll
            for (int s = 0; s < 8; ++s)
                acc = __builtin_amdgcn_wmma_f32_16x16x4_f32(
                    false, af[s], false, bf[nt][s], (short)0, acc, false, false);
            #pragma unroll
            for (int e = 0; e < 8; ++e) {
                float h = fmaxf(acc[e] + cb[nt], 0.f);
                lw[(e + 8*hl)*64 + 16*nt + ln] = h;   // rows x 64 feats
            }
        }
        // wait for this wave's LDS writes (cross-lane hazard), then tail
        asm volatile("s_wait_dscnt 0" ::: "memory");
        if (hl == 0) {
            float o0 = f2b[0], o1 = f2b[1], o2 = f2b[2];
            #pragma unroll 8
            for (int n = 0; n < 64; ++n) {
                float hv = lw[ln*64 + n];
                o0 = fmaf(hv, f2w[n],      o0);
                o1 = fmaf(hv, f2w[64+n],   o1);
                o2 = fmaf(hv, f2w[128+n],  o2);
            }
            out[row*3+0] = o0; out[row*3+1] = o1; out[row*3+2] = o2;
        }
    }
}

// ---------------------------------------------------------------------------
extern "C" void kernel_launch(void* const* d_in, const int* in_sizes, int n_in,
                              void* d_out, int out_size, void* d_ws, size_t ws_size,
                              hipStream_t stream) {
    const float* x          = (const float*)d_in[0];
    const float* ehr_w      = (const float*)d_in[1];
    const float* ehr_b      = (const float*)d_in[2];
    // d_in[3], d_in[4]: img_w/img_b -> dead code in the reference graph
    const float* bio_w      = (const float*)d_in[5];
    const float* bio_b      = (const float*)d_in[6];
    const float* bio_qkv_w  = (const float*)d_in[7];
    const float* bio_qkv_b  = (const float*)d_in[8];
    const float* ehr_qkv_w  = (const float*)d_in[9];
    const float* ehr_qkv_b  = (const float*)d_in[10];
    // d_in[11], d_in[12]: img_qkv_* -> dead code
    const float* attn_in_w  = (const float*)d_in[13];
    const float* attn_in_b  = (const float*)d_in[14];
    const float* attn_out_w = (const float*)d_in[15];
    const float* attn_out_b = (const float*)d_in[16];
    const float* ab_proj_w  = (const float*)d_in[17];
    const float* ab_proj_b  = (const float*)d_in[18];
    const float* f1_w       = (const float*)d_in[19];
    const float* f1_b       = (const float*)d_in[20];
    const float* bn_g       = (const float*)d_in[21];
    const float* bn_b       = (const float*)d_in[22];
    const float* f2_w       = (const float*)d_in[23];
    const float* f2_b       = (const float*)d_in[24];
    float* ws  = (float*)d_ws;
    float* out = (float*)d_out;

    const int B = in_sizes[0] / 3;     // 1048576, divisible by 16
    const int ntiles = B / 16;

    prep_kernel<<<1, 64, 0, stream>>>(ehr_qkv_w, ehr_qkv_b, bio_qkv_w, bio_qkv_b,
                                      attn_in_w, attn_in_b, attn_out_w, attn_out_b,
                                      ab_proj_w, ab_proj_b, f1_w, f1_b, ws);
    pass1_kernel<<<1024, 256, 0, stream>>>(x, ehr_w, ehr_b, bio_w, bio_b, ws, ntiles);
    fold_kernel<<<1, 64, 0, stream>>>(bn_g, bn_b, ws, 1.0f / (float)B);
    pass2_kernel<<<1024, 256, 0, stream>>>(x, ehr_w, ehr_b, bio_w, bio_b,
                                           f2_w, f2_b, ws, out, ntiles);
}